// SparseMemory_25486335935179
// MI455X (gfx1250) — compile-verified
//
#include <hip/hip_runtime.h>
#include <math.h>

// ---------------- problem constants ----------------
#define B_   32
#define S_   2
#define D_   512
#define M_   32768
#define CW_  64
#define R_   8
#define K_   4
#define N_   66        // S*(K*R+1)
#define E_   578       // CW*R + CW + 2
#define NQ   16        // S*R query vectors per batch
#define DELTA_ 0.005f
#define EPS_   1e-6f
#define CHUNK  2048
#define NCHUNK 16      // M_/CHUNK

typedef __attribute__((ext_vector_type(2))) float v2f;
typedef __attribute__((ext_vector_type(8))) float v8f;

// ---------------- workspace layout (bytes) ----------------
constexpr size_t OFF_QN     = 0;                                   // B*NQ*CW f32 (normalized queries)
constexpr size_t OFF_WVEC   = OFF_QN     + (size_t)B_*NQ*CW_*4;    // B*S*CW f32
constexpr size_t OFF_IW     = OFF_WVEC   + (size_t)B_*S_*CW_*4;    // B*S*2 f32 (ig,wg)
constexpr size_t OFF_RELNEW = OFF_IW     + (size_t)B_*S_*2*4;      // B*N f32
constexpr size_t OFF_NV     = OFF_RELNEW + (size_t)B_*N_*4;        // B*N*CW f32 (new visible rows)
constexpr size_t OFF_PV     = OFF_NV     + (size_t)B_*N_*CW_*4;    // B*NCHUNK*NQ*4 f32
constexpr size_t OFF_PI     = OFF_PV     + (size_t)B_*NCHUNK*NQ*4*4;
constexpr size_t OFF_LUM    = OFF_PI     + (size_t)B_*NCHUNK*NQ*4*4; // B*S int
constexpr size_t OFF_NEWPOS = OFF_LUM    + (size_t)B_*S_*4;        // B*N int
constexpr size_t OFF_IDXMAP = OFF_NEWPOS + (size_t)B_*N_*4;        // B*M int

// jax top_k order: value descending, ties -> smaller index first
__device__ __forceinline__ bool tk_better(float s, int m, float v, int i) {
  return s > v || (s == v && m < i);
}
// branch-free sorted-4 insert on scalar registers (no alloca -> no scratch)
#define TOP4_PUSH(s, m, v0, v1, v2, v3, i0, i1, i2, i3)                         \
  do {                                                                          \
    float _s = (s); int _m = (m);                                               \
    if (tk_better(_s, _m, v3, i3)) { v3 = _s; i3 = _m; }                        \
    if (tk_better(v3, i3, v2, i2)) { float tf=v2; v2=v3; v3=tf; int tq=i2; i2=i3; i3=tq; } \
    if (tk_better(v2, i2, v1, i1)) { float tf=v1; v1=v2; v2=tf; int tq=i1; i1=i2; i2=tq; } \
    if (tk_better(v1, i1, v0, i0)) { float tf=v0; v0=v1; v1=tf; int tq=i0; i0=i1; i1=tq; } \
  } while (0)

// ---------------- K0: idx_map = -1 ----------------
__global__ void init_idxmap_kernel(int* __restrict__ idx_map) {
  size_t i = (size_t)blockIdx.x * blockDim.x + threadIdx.x;
  const size_t total = (size_t)B_ * M_;
  const size_t stride = (size_t)gridDim.x * blockDim.x;
  for (; i < total; i += stride) idx_map[i] = -1;
}

// ---------------- K1: xi = x @ W^T + bW ; derived quantities ----------------
__global__ __launch_bounds__(128) void proj_kernel(
    const float* __restrict__ x, const float* __restrict__ W, const float* __restrict__ bW,
    float* __restrict__ qn, float* __restrict__ wvec, float* __restrict__ iw)
{
  __shared__ __align__(16) float xrow[D_];
  __shared__ float xi[E_];
  __shared__ float rq[R_];
  const int bs = blockIdx.x;   // b*S + s
  const int t  = threadIdx.x;
  const float* xr = x + (size_t)bs * D_;
  for (int d = t; d < D_; d += 128) xrow[d] = xr[d];
  __syncthreads();
  for (int e = t; e < E_; e += 128) {
    const float* wr = W + (size_t)e * D_;
    float acc = 0.f;
    for (int d = 0; d < D_; ++d) acc += xrow[d] * wr[d];
    xi[e] = acc + bW[e];
  }
  __syncthreads();
  if (t < R_) {
    float s = 0.f;
#pragma unroll 8
    for (int k = 0; k < CW_; ++k) { float v = xi[t*CW_ + k]; s += v*v; }
    rq[t] = 1.0f / (sqrtf(s) + EPS_);
  }
  __syncthreads();
  const int b = bs / S_, s = bs % S_;
  for (int e = t; e < R_*CW_; e += 128) {
    int r = e >> 6, w = e & 63;
    qn[((size_t)b*NQ + s*R_ + r)*CW_ + w] = xi[e] * rq[r];
  }
  if (t < CW_) wvec[(size_t)bs*CW_ + t] = xi[R_*CW_ + t];
  if (t == 0) {
    iw[bs*2 + 0] = 1.0f / (1.0f + expf(-xi[E_-2]));   // ig
    iw[bs*2 + 1] = 1.0f / (1.0f + expf(-xi[E_-1]));   // wg
  }
}

// ---------------- K2: write phase (per batch) ----------------
__global__ __launch_bounds__(128) void write_kernel(
    const float* __restrict__ visible_memory, const float* __restrict__ read_weights,
    const float* __restrict__ usage, const int* __restrict__ read_positions,
    const int* __restrict__ timestep,
    const float* __restrict__ wvec, const float* __restrict__ iw,
    float* __restrict__ rel_new, float* __restrict__ new_vis, int* __restrict__ idx_map)
{
  __shared__ int   pos[N_];
  __shared__ float rel[N_];
  __shared__ float ww[S_][N_];
  __shared__ float er[N_];
  __shared__ float wv[S_][CW_];
  __shared__ float minus[2];
  const int b = blockIdx.x;
  const int t = threadIdx.x;
  if (t < N_) {
    pos[t] = read_positions[b*N_ + t];
    rel[t] = usage[(size_t)b*M_ + pos[t]];
  }
  if (t < S_*CW_) wv[t>>6][t&63] = wvec[(size_t)b*S_*CW_ + t];
  __syncthreads();
  if (t == 0) {
    float v0 = INFINITY, v1 = INFINITY;
    for (int n = 0; n < N_; ++n) {
      float u = rel[n];
      if (u < v0) { v1 = v0; v0 = u; }
      else if (u < v1) { v1 = u; }
    }
    minus[0] = v0; minus[1] = v1;
  }
  __syncthreads();
  if (t < N_) {
    float I0 = (rel[t] == minus[0]) ? 1.f : 0.f;
    float I1 = (rel[t] == minus[1]) ? 1.f : 0.f;
    float ig0 = iw[(b*S_+0)*2+0], wg0 = iw[(b*S_+0)*2+1];
    float ig1 = iw[(b*S_+1)*2+0], wg1 = iw[(b*S_+1)*2+1];
    float rw0 = read_weights[((size_t)b*S_+0)*N_ + t];
    float rw1 = read_weights[((size_t)b*S_+1)*N_ + t];
    float w0 = wg0 * (ig0*rw0 + (1.f-ig0)*I0);
    float w1 = wg1 * (ig1*rw1 + (1.f-ig1)*I1);
    ww[0][t] = w0; ww[1][t] = w1;
    float u = (fabsf(rw0+rw1) + fabsf(w0+w1) > DELTA_) ? 1.f : 0.f;
    float ts = (float)(*timestep);
    rel_new[b*N_ + t] = ts*u + rel[t]*(1.f - u);
    er[t] = (I0 + I1 >= 1.f) ? 1.f : 0.f;
  }
  __syncthreads();
  for (int e = t; e < N_*CW_; e += 128) {
    int n = e >> 6, w = e & 63;
    float nv = visible_memory[((size_t)b*N_ + n)*CW_ + w] * (1.f - er[n])
             + ww[0][n]*wv[0][w] + ww[1][n]*wv[1][w];
    new_vis[((size_t)b*N_ + n)*CW_ + w] = nv;
  }
  if (t == 0) {   // serial scatter: deterministic "last write wins" for duplicates
    for (int n = 0; n < N_; ++n) idx_map[(size_t)b*M_ + pos[n]] = n;
  }
}

// ---------------- K_lum: 2 smallest updated-usage indices per batch ----------------
__global__ __launch_bounds__(256) void lum_kernel(
    const float* __restrict__ usage, const int* __restrict__ idx_map,
    const float* __restrict__ rel_new, int* __restrict__ lum)
{
  __shared__ float sv[256][2];
  __shared__ int   si[256][2];
  const int b = blockIdx.x;
  const int t = threadIdx.x;
  float v0 = INFINITY, v1 = INFINITY; int i0 = 0x7fffffff, i1 = 0x7fffffff;
  for (int m = t; m < M_; m += 256) {
    float u = usage[(size_t)b*M_ + m];
    int vi = idx_map[(size_t)b*M_ + m];
    if (vi >= 0) u = rel_new[b*N_ + vi];
    if (u < v0 || (u == v0 && m < i0)) { v1=v0; i1=i0; v0=u; i0=m; }
    else if (u < v1 || (u == v1 && m < i1)) { v1=u; i1=m; }
  }
  sv[t][0]=v0; sv[t][1]=v1; si[t][0]=i0; si[t][1]=i1;
  __syncthreads();
  if (t == 0) {
    float b0=INFINITY,b1=INFINITY; int j0=0x7fffffff,j1=0x7fffffff;
    for (int k = 0; k < 256; ++k)
      for (int h = 0; h < 2; ++h) {
        float u = sv[k][h]; int i = si[k][h];
        if (u < b0 || (u == b0 && i < j0)) { b1=b0; j1=j0; b0=u; j0=i; }
        else if (u < b1 || (u == b1 && i < j1)) { b1=u; j1=i; }
      }
    lum[b*2+0]=j0; lum[b*2+1]=j1;
  }
}

// ---------------- K3 helpers ----------------
__device__ __forceinline__ void fetch_tile(float4 nb[8],
    const float* __restrict__ memory, const float* __restrict__ new_vis,
    const int* __restrict__ idx_map, int b, int m, int half)
{
  const int vi = idx_map[(size_t)b*M_ + m];
  const float* src = (vi >= 0)
      ? (new_vis + ((size_t)b*N_ + vi)*CW_)
      : (memory  + ((size_t)b*M_ + m)*CW_);
  const float4* s4 = (const float4*)(src + half*32);
#pragma unroll
  for (int i = 0; i < 8; ++i) nb[i] = s4[i];
}

// store prefetched half-row to LDS and produce rinv from registers (+1 shuffle)
__device__ __forceinline__ void commit_tile(const float4 nb[8],
    float* __restrict__ rowdst, float* __restrict__ rinv_slot, int lane)
{
  float4* d4 = (float4*)rowdst;
  float s = 0.f;
#pragma unroll
  for (int i = 0; i < 8; ++i) {
    d4[i] = nb[i];
    s += nb[i].x*nb[i].x + nb[i].y*nb[i].y + nb[i].z*nb[i].z + nb[i].w*nb[i].w;
  }
  float tot = s + __shfl_xor(s, 1, 32);
  if ((lane & 1) == 0) *rinv_slot = 1.0f / (sqrtf(tot) + EPS_);
}

// ---------------- K3: f32 WMMA similarity + running top-4 ----------------
// grid = (NCHUNK, B_), 256 threads = 8 waves; each wave owns its own tiles,
// double-buffered LDS + register prefetch; wave-private phases use only
// compiler fences (intra-wave LDS is in-order per ISA §7.3).
__global__ __launch_bounds__(256) void sim_topk_kernel(
    const float* __restrict__ memory, const float* __restrict__ qn,
    const int* __restrict__ idx_map, const float* __restrict__ new_vis,
    float* __restrict__ pvals, int* __restrict__ pidx)
{
  __shared__ __align__(16) float rows[8][2][16][CW_];   // 64 KB, double-buffered
  __shared__ float rinv[8][2][16];
  __shared__ float scores[8][NQ][16];
  __shared__ float mergeV[8][NQ][4];
  __shared__ int   mergeI[8][NQ][4];

  const int b     = blockIdx.y;
  const int chunk = blockIdx.x;
  const int w     = threadIdx.x >> 5;
  const int lane  = threadIdx.x & 31;
  const int hi    = lane >> 4;
  const int qcol  = lane & 15;          // A: query row / B: memory col / D: n
  const int c0    = hi ? 2 : 0;         // K offset within 4-wide step
  const int lrow  = lane >> 1;          // row this lane loads
  const int lhalf = lane & 1;

  // A fragments (fixed for whole kernel): query = qcol, K pair {c0,c0+1}
  v2f afrag[16];
  const float* qbase = qn + ((size_t)b*NQ + qcol)*CW_;
#pragma unroll
  for (int kk = 0; kk < 16; ++kk)
    afrag[kk] = *(const v2f*)(qbase + kk*4 + c0);

  // running top-4 in scalar registers; lanes 0..15 own query==lane
  float tv0=-INFINITY, tv1=-INFINITY, tv2=-INFINITY, tv3=-INFINITY;
  int   ti0=0x7fffffff, ti1=0x7fffffff, ti2=0x7fffffff, ti3=0x7fffffff;

  const int m0 = chunk * CHUNK;
  const int T  = CHUNK / 16 / 8;        // tiles per wave
  float4 nb[8];

  fetch_tile(nb, memory, new_vis, idx_map, b, m0 + w*16 + lrow, lhalf);
  commit_tile(nb, &rows[w][0][lrow][lhalf*32], &rinv[w][0][lrow], lane);

  for (int it = 0; it < T; ++it) {
    const int cur = it & 1;
    const int mbase = m0 + (w + it*8)*16;
    const bool havenext = (it + 1 < T);
    if (havenext)   // issue next tile's global loads now; they overlap the WMMAs
      fetch_tile(nb, memory, new_vis, idx_map, b, m0 + (w + (it+1)*8)*16 + lrow, lhalf);
    __builtin_amdgcn_wave_barrier();

    // 16 x V_WMMA_F32_16X16X4_F32 -> full 64-dim dots, f32 precision
    v8f acc = {};
#pragma unroll
    for (int kk = 0; kk < 16; ++kk) {
      v2f bfrag = *(const v2f*)(&rows[w][cur][qcol][kk*4 + c0]);
      acc = __builtin_amdgcn_wmma_f32_16x16x4_f32(
          false, afrag[kk], false, bfrag, (short)0, acc, false, false);
    }
    const float rv = rinv[w][cur][qcol];   // column scale = 1/(||row||+eps)
#pragma unroll
    for (int v = 0; v < 8; ++v)
      scores[w][v + hi*8][qcol] = acc[v] * rv;
    __builtin_amdgcn_wave_barrier();

    if (lane < 16) {
#pragma unroll
      for (int n = 0; n < 16; ++n)
        TOP4_PUSH(scores[w][lane][n], mbase + n, tv0,tv1,tv2,tv3, ti0,ti1,ti2,ti3);
    }
    __builtin_amdgcn_wave_barrier();

    if (havenext)
      commit_tile(nb, &rows[w][cur^1][lrow][lhalf*32], &rinv[w][cur^1][lrow], lane);
    __builtin_amdgcn_wave_barrier();
  }

  // cross-wave merge within workgroup (real barrier needed here)
  if (lane < 16) {
    mergeV[w][lane][0]=tv0; mergeV[w][lane][1]=tv1; mergeV[w][lane][2]=tv2; mergeV[w][lane][3]=tv3;
    mergeI[w][lane][0]=ti0; mergeI[w][lane][1]=ti1; mergeI[w][lane][2]=ti2; mergeI[w][lane][3]=ti3;
  }
  __syncthreads();
  if (threadIdx.x < NQ) {
    const int q = threadIdx.x;
    float v0=-INFINITY, v1=-INFINITY, v2=-INFINITY, v3=-INFINITY;
    int   j0=0x7fffffff, j1=0x7fffffff, j2=0x7fffffff, j3=0x7fffffff;
    for (int w2 = 0; w2 < 8; ++w2)
      for (int k = 0; k < 4; ++k)
        TOP4_PUSH(mergeV[w2][q][k], mergeI[w2][q][k], v0,v1,v2,v3, j0,j1,j2,j3);
    const size_t base = (((size_t)b*NCHUNK + chunk)*NQ + q)*4;
    pvals[base+0]=v0; pvals[base+1]=v1; pvals[base+2]=v2; pvals[base+3]=v3;
    pidx [base+0]=j0; pidx [base+1]=j1; pidx [base+2]=j2; pidx [base+3]=j3;
  }
}

// ---------------- K4: merge chunk partials -> new_pos ----------------
__global__ __launch_bounds__(64) void merge_kernel(
    const float* __restrict__ pvals, const int* __restrict__ pidx,
    const int* __restrict__ lum, int* __restrict__ new_pos)
{
  const int b = blockIdx.x;
  const int t = threadIdx.x;
  if (t < NQ) {
    float v0=-INFINITY, v1=-INFINITY, v2=-INFINITY, v3=-INFINITY;
    int   j0=0x7fffffff, j1=0x7fffffff, j2=0x7fffffff, j3=0x7fffffff;
    for (int c = 0; c < NCHUNK; ++c) {           // increasing index ranges -> tie order correct
      const size_t base = (((size_t)b*NCHUNK + c)*NQ + t)*4;
      for (int k = 0; k < 4; ++k)
        TOP4_PUSH(pvals[base+k], pidx[base+k], v0,v1,v2,v3, j0,j1,j2,j3);
    }
    const int s = t / R_, r = t % R_;
    const int o = b*N_ + s*(K_*R_+1) + r*K_;
    new_pos[o+0]=j0; new_pos[o+1]=j1; new_pos[o+2]=j2; new_pos[o+3]=j3;
  }
  if (t == 0) {
    new_pos[b*N_ + 0*(K_*R_+1) + K_*R_] = lum[b*2+0];
    new_pos[b*N_ + 1*(K_*R_+1) + K_*R_] = lum[b*2+1];
  }
}

// ---------------- K5: gather vis, sim2, softmax, read_vectors ----------------
__global__ __launch_bounds__(128) void read_kernel(
    const float* __restrict__ memory, const float* __restrict__ qn,
    const int* __restrict__ idx_map, const float* __restrict__ new_vis,
    const int* __restrict__ new_pos, float* __restrict__ out)
{
  __shared__ __align__(16) float vis[N_][CW_];
  __shared__ float rinvv[N_];
  __shared__ float qnl[NQ][CW_];
  __shared__ float rw[NQ][N_];
  const int b = blockIdx.x;
  const int t = threadIdx.x;
  for (int n = t; n < N_; n += 128) {
    const int p  = new_pos[b*N_ + n];
    const int vi = idx_map[(size_t)b*M_ + p];
    const float* src = (vi >= 0) ? (new_vis + ((size_t)b*N_ + vi)*CW_)
                                 : (memory  + ((size_t)b*M_ + p)*CW_);
#pragma unroll
    for (int k = 0; k < CW_; k += 4) *(float4*)&vis[n][k] = *(const float4*)(src + k);
  }
  for (int e = t; e < NQ*CW_; e += 128) qnl[e>>6][e&63] = qn[(size_t)b*NQ*CW_ + e];
  __syncthreads();
  if (t < N_) {
    float s = 0.f;
#pragma unroll 8
    for (int k = 0; k < CW_; ++k) { float v = vis[t][k]; s += v*v; }
    rinvv[t] = 1.0f / (sqrtf(s) + EPS_);
  }
  __syncthreads();
  for (int e = t; e < NQ*N_; e += 128) {
    const int q = e / N_, n = e % N_;
    float s = 0.f;
#pragma unroll 8
    for (int k = 0; k < CW_; ++k) s += qnl[q][k] * vis[n][k];
    rw[q][n] = s * rinvv[n];
  }
  __syncthreads();
  if (t < NQ) {
    float mx = -INFINITY;
    for (int n = 0; n < N_; ++n) mx = fmaxf(mx, rw[t][n]);
    float sum = 0.f;
    for (int n = 0; n < N_; ++n) { float e2 = expf(rw[t][n] - mx); rw[t][n] = e2; sum += e2; }
    const float inv = 1.0f / sum;
    for (int n = 0; n < N_; ++n) rw[t][n] *= inv;
  }
  __syncthreads();
  for (int e = t; e < R_*CW_; e += 128) {
    const int r = e >> 6, k = e & 63;
    float acc = 0.f;
    for (int n = 0; n < N_; ++n)
      acc += (rw[r][n] + rw[R_ + r][n]) * vis[n][k];   // sums over s and n
    out[((size_t)b*R_ + r)*CW_ + k] = acc;
  }
}

// ---------------- host launcher ----------------
extern "C" void kernel_launch(void* const* d_in, const int* in_sizes, int n_in,
                              void* d_out, int out_size, void* d_ws, size_t ws_size,
                              hipStream_t stream)
{
  const float* x               = (const float*)d_in[0];
  const float* W               = (const float*)d_in[1];
  const float* bW              = (const float*)d_in[2];
  const float* memory          = (const float*)d_in[3];
  const float* visible_memory  = (const float*)d_in[4];
  const float* read_weights    = (const float*)d_in[5];
  const float* usage           = (const float*)d_in[6];
  const int*   read_positions  = (const int*)d_in[7];
  // d_in[8] least_used_mem: unused by the reference math
  const int*   timestep        = (const int*)d_in[9];

  char* ws = (char*)d_ws;
  float* qn      = (float*)(ws + OFF_QN);
  float* wvec    = (float*)(ws + OFF_WVEC);
  float* iw      = (float*)(ws + OFF_IW);
  float* rel_new = (float*)(ws + OFF_RELNEW);
  float* new_vis = (float*)(ws + OFF_NV);
  float* pvals   = (float*)(ws + OFF_PV);
  int*   pidx    = (int*)  (ws + OFF_PI);
  int*   lum     = (int*)  (ws + OFF_LUM);
  int*   new_pos = (int*)  (ws + OFF_NEWPOS);
  int*   idx_map = (int*)  (ws + OFF_IDXMAP);

  init_idxmap_kernel<<<1024, 256, 0, stream>>>(idx_map);
  proj_kernel<<<B_*S_, 128, 0, stream>>>(x, W, bW, qn, wvec, iw);
  write_kernel<<<B_, 128, 0, stream>>>(visible_memory, read_weights, usage,
                                       read_positions, timestep, wvec, iw,
                                       rel_new, new_vis, idx_map);
  lum_kernel<<<B_, 256, 0, stream>>>(usage, idx_map, rel_new, lum);
  dim3 g3(NCHUNK, B_);
  sim_topk_kernel<<<g3, 256, 0, stream>>>(memory, qn, idx_map, new_vis, pvals, pidx);
  merge_kernel<<<B_, 64, 0, stream>>>(pvals, pidx, lum, new_pos);
  read_kernel<<<B_, 128, 0, stream>>>(memory, qn, idx_map, new_vis, new_pos, (float*)d_out);
}